// HierarchicalClassifier_66769561584338
// MI455X (gfx1250) — compile-verified
//
#include <hip/hip_runtime.h>
#include <math.h>

typedef __attribute__((ext_vector_type(2))) float v2f;
typedef __attribute__((ext_vector_type(8))) float v8f;

// Async global->LDS staging via inline asm (the clang builtins for these use
// HIP LangAS-qualified pointer params that can't be spelled from HIP source).
#if defined(__gfx1250__)
#define USE_ASYNC_LDS 1
#else
#define USE_ASYNC_LDS 0
#endif

// Problem constants
constexpr int B = 8192, D = 2048, E = 16, P = 1024, C0 = 64, C1 = 128;
constexpr float EPS = 1e-5f;

// Flattened output offsets (return order: plogits, cl0, cl1, pproj, cp0, cp1)
constexpr size_t OFF_PLOG  = 0;
constexpr size_t OFF_CL0   = OFF_PLOG + (size_t)B * E;       // 131072
constexpr size_t OFF_CL1   = OFF_CL0  + (size_t)B * C0;      // 655360
constexpr size_t OFF_PPROJ = OFF_CL1  + (size_t)B * C1;      // 1703936
constexpr size_t OFF_CP0   = OFF_PPROJ + (size_t)B * P;      // 10092544
constexpr size_t OFF_CP1   = OFF_CP0   + (size_t)B * P;      // 18481152

constexpr int KCH = 512;          // K chunk staged in LDS
constexpr int XSS = KCH + 4;      // padded LDS row stride (bank-conflict-free A loads)
constexpr int NCH = D / KCH;      // chunks per fc GEMM
static_assert(D % KCH == 0, "");

__global__ void init_cnt_kernel(int* __restrict__ cnt) {
    if (threadIdx.x < E) cnt[threadIdx.x] = 0;
}

// Stage one 16 x KCH f32 tile of gathered x rows into LDS.
// Async path: per-lane GLOBAL_LOAD_ASYNC_TO_LDS_B128 (tracked by ASYNCcnt).
template <class RowMap>
__device__ __forceinline__ void stage_chunk(float* __restrict__ dst,
                                            const float* __restrict__ x,
                                            RowMap rowOf, int k0, int tid)
{
#if USE_ASYNC_LDS
    for (int idx4 = tid; idx4 < 16 * (KCH / 4); idx4 += 256) {
        const int r  = idx4 >> 7;              // KCH/4 = 128 b128's per row
        const int c4 = idx4 & (KCH / 4 - 1);
        const unsigned long long gaddr =
            (unsigned long long)(uintptr_t)(x + (size_t)rowOf(r) * D + k0 + c4 * 4);
        // Low 32 bits of a flat shared pointer are the LDS byte offset.
        const unsigned lofs = (unsigned)(uintptr_t)(dst + r * XSS + c4 * 4);
        asm volatile("global_load_async_to_lds_b128 %0, %1, off"
                     :: "v"(lofs), "v"(gaddr) : "memory");
    }
#else
    for (int idx = tid; idx < 16 * KCH; idx += 256)
        dst[(idx >> 9) * XSS + (idx & (KCH - 1))] =
            x[(size_t)rowOf(idx >> 9) * D + k0 + (idx & (KCH - 1))];
#endif
}

__device__ __forceinline__ void wait_stage() {
#if USE_ASYNC_LDS
    asm volatile("s_wait_asynccnt 0" ::: "memory");
#endif
    __syncthreads();
}

// ---------------------------------------------------------------------------
// Parent: fc [16xD]x[DxE] -> LN -> argmax (counting sort) -> proj [16xE]x[ExP]
// ---------------------------------------------------------------------------
__global__ __launch_bounds__(256) void parent_kernel(
    const float* __restrict__ x,
    const float* __restrict__ pwfc,  const float* __restrict__ pbfc,
    const float* __restrict__ pwproj,const float* __restrict__ pbproj,
    float* __restrict__ out, int* __restrict__ cnt, int* __restrict__ rowlist)
{
    __shared__ __align__(16) float xs[2][16 * XSS];   // 66 KB double-buffered x tile
    __shared__ float partial[8][256];                 // per-wave partial 16x16 tiles
    __shared__ float lg[16 * 17];                     // logits / normalized logits

    const int tid  = threadIdx.x;
    const int wave = tid >> 5;
    const int lane = tid & 31;
    const int mrow = lane & 15;
    const int koff = (lane >> 4) * 2;
    const int rowbase = blockIdx.x * 16;
    auto rowmap = [&](int r) { return rowbase + r; };

    // -------- fc: acc[m][e] = sum_k x[m][k] * pwfc[e][k]; K split over waves,
    //          x chunks double-buffered through LDS via async copies.
    v8f acc = {};
    const float* bsrc = pwfc + (size_t)mrow * D;      // N index (expert) = lane&15
    stage_chunk(xs[0], x, rowmap, 0, tid);
    wait_stage();
    for (int c = 0; c < NCH; ++c) {
        if (c + 1 < NCH) stage_chunk(xs[(c + 1) & 1], x, rowmap, (c + 1) * KCH, tid);
        const float* xb = xs[c & 1];
        const int kbeg = wave * (KCH / 8);
        for (int k = kbeg; k < kbeg + KCH / 8; k += 4) {
            v2f a, b;
            a.x = xb[mrow * XSS + k + koff];
            a.y = xb[mrow * XSS + k + koff + 1];
            b.x = bsrc[c * KCH + k + koff];
            b.y = bsrc[c * KCH + k + koff + 1];
            acc = __builtin_amdgcn_wmma_f32_16x16x4_f32(false, a, false, b,
                                                        (short)0, acc, false, false);
        }
        wait_stage();
    }
#pragma unroll
    for (int v = 0; v < 8; ++v)
        partial[wave][(v + 8 * (lane >> 4)) * 16 + mrow] = acc[v];
    __syncthreads();
    {   // reduce 8 partials; tid encodes (m = tid>>4, n = tid&15)
        float s = pbfc[tid & 15];
#pragma unroll
        for (int w = 0; w < 8; ++w) s += partial[w][tid];
        lg[(tid >> 4) * 17 + (tid & 15)] = s;
    }
    __syncthreads();

    // -------- LN + argmax + counting sort (one thread per row)
    if (tid < 16) {
        float mu = 0.f;
        for (int n = 0; n < 16; ++n) mu += lg[tid * 17 + n];
        mu *= (1.f / 16.f);
        float var = 0.f;
        for (int n = 0; n < 16; ++n) { float d = lg[tid * 17 + n] - mu; var += d * d; }
        var *= (1.f / 16.f);
        const float rs = rsqrtf(var + EPS);
        float best = -INFINITY; int bi = 0;
        for (int n = 0; n < 16; ++n) {
            float h = (lg[tid * 17 + n] - mu) * rs;
            lg[tid * 17 + n] = h;
            out[OFF_PLOG + (size_t)(rowbase + tid) * E + n] = h;
            if (h > best) { best = h; bi = n; }   // strict '>' keeps first max
        }
        int pos = atomicAdd(&cnt[bi], 1);
        rowlist[bi * B + pos] = rowbase + tid;
    }
    __syncthreads();

    // -------- proj: [16x16] x [16xP], K = 16; 8 N-tiles per wave
    for (int t = 0; t < 8; ++t) {
        const int n0 = (wave * 8 + t) * 16;
        v8f pacc = {};
        const float* wp = pwproj + (size_t)(n0 + mrow) * E;
#pragma unroll
        for (int kc = 0; kc < 16; kc += 4) {
            v2f a, b;
            a.x = lg[mrow * 17 + kc + koff];
            a.y = lg[mrow * 17 + kc + koff + 1];
            b.x = wp[kc + koff];
            b.y = wp[kc + koff + 1];
            pacc = __builtin_amdgcn_wmma_f32_16x16x4_f32(false, a, false, b,
                                                         (short)0, pacc, false, false);
        }
        const float bias = pbproj[n0 + mrow];
#pragma unroll
        for (int v = 0; v < 8; ++v) {
            const int m = v + 8 * (lane >> 4);
            out[OFF_PPROJ + (size_t)(rowbase + m) * P + n0 + mrow] = pacc[v] + bias;
        }
    }
}

// ---------------------------------------------------------------------------
// Child level: grouped GEMM over the per-expert row lists built by parent.
//   fc:   [16xD] x [DxC]  (one 16-wide N-tile per wave)
//   LN over C, scatter cl; proj: [16xC] x [CxP] (64 N-tiles, 8 per wave)
// ---------------------------------------------------------------------------
template <int C>
__global__ __launch_bounds__(256) void child_kernel(
    const float* __restrict__ x,
    const float* __restrict__ wfc,  const float* __restrict__ bfc,
    const float* __restrict__ wproj,const float* __restrict__ bproj,
    const int* __restrict__ cnt, const int* __restrict__ rowlist,
    float* __restrict__ cl_out, float* __restrict__ cp_out)
{
    constexpr int NT = C / 16;                        // fc N-tiles (4 or 8)
    __shared__ __align__(16) float xs[2][16 * XSS];   // 66 KB double-buffered gather
    __shared__ float sel[16][C + 4];                  // child logits / normalized h
    __shared__ int   rows_s[16];

    const int e     = blockIdx.y;
    const int tileb = blockIdx.x * 16;
    const int total = cnt[e];
    if (tileb >= total) return;                       // uniform across block
    const int nvalid = min(16, total - tileb);

    const int tid  = threadIdx.x;
    const int wave = tid >> 5;
    const int lane = tid & 31;
    const int mrow = lane & 15;
    const int koff = (lane >> 4) * 2;

    if (tid < 16)
        rows_s[tid] = rowlist[e * B + tileb + ((tid < nvalid) ? tid : 0)];
    __syncthreads();
    auto rowmap = [&](int r) { return rows_s[r]; };

    // -------- fc with double-buffered async-staged x chunks
    v8f acc = {};
    const int wv = (wave < NT) ? wave : 0;
    const float* wrow = wfc + ((size_t)e * C + wv * 16 + mrow) * D;
    stage_chunk(xs[0], x, rowmap, 0, tid);
    wait_stage();
    for (int c = 0; c < NCH; ++c) {
        if (c + 1 < NCH) stage_chunk(xs[(c + 1) & 1], x, rowmap, (c + 1) * KCH, tid);
        if (wave < NT) {
            const float* xb = xs[c & 1];
            for (int k = 0; k < KCH; k += 4) {
                v2f a, b;
                a.x = xb[mrow * XSS + k + koff];
                a.y = xb[mrow * XSS + k + koff + 1];
                b.x = wrow[c * KCH + k + koff];
                b.y = wrow[c * KCH + k + koff + 1];
                acc = __builtin_amdgcn_wmma_f32_16x16x4_f32(false, a, false, b,
                                                            (short)0, acc, false, false);
            }
        }
        wait_stage();
    }
    if (wave < NT) {
        const float bias = bfc[e * C + wave * 16 + mrow];
#pragma unroll
        for (int v = 0; v < 8; ++v)
            sel[v + 8 * (lane >> 4)][wave * 16 + mrow] = acc[v] + bias;
    }
    __syncthreads();

    // -------- LN per row
    if (tid < 16) {
        float mu = 0.f;
        for (int c = 0; c < C; ++c) mu += sel[tid][c];
        mu *= (1.f / C);
        float var = 0.f;
        for (int c = 0; c < C; ++c) { float d = sel[tid][c] - mu; var += d * d; }
        var *= (1.f / C);
        const float rs = rsqrtf(var + EPS);
        for (int c = 0; c < C; ++c) sel[tid][c] = (sel[tid][c] - mu) * rs;
    }
    __syncthreads();

    // scatter child logits
    for (int idx = tid; idx < 16 * C; idx += 256) {
        const int r = idx / C, c = idx % C;
        if (r < nvalid) cl_out[(size_t)rows_s[r] * C + c] = sel[r][c];
    }

    // -------- proj: K = C, N = P (64 tiles; 8 per wave)
    for (int t = 0; t < 8; ++t) {
        const int n0 = (wave * 8 + t) * 16;
        v8f pacc = {};
        const float* wp = wproj + ((size_t)e * P + n0 + mrow) * C;
        for (int kc = 0; kc < C; kc += 4) {
            v2f a, b;
            a.x = sel[mrow][kc + koff];
            a.y = sel[mrow][kc + koff + 1];
            b.x = wp[kc + koff];
            b.y = wp[kc + koff + 1];
            pacc = __builtin_amdgcn_wmma_f32_16x16x4_f32(false, a, false, b,
                                                         (short)0, pacc, false, false);
        }
        const float bias = bproj[e * P + n0 + mrow];
#pragma unroll
        for (int v = 0; v < 8; ++v) {
            const int m = v + 8 * (lane >> 4);
            if (m < nvalid)
                cp_out[(size_t)rows_s[m] * P + n0 + mrow] = pacc[v] + bias;
        }
    }
}

// ---------------------------------------------------------------------------
extern "C" void kernel_launch(void* const* d_in, const int* in_sizes, int n_in,
                              void* d_out, int out_size, void* d_ws, size_t ws_size,
                              hipStream_t stream)
{
    const float* x      = (const float*)d_in[0];
    const float* pwfc   = (const float*)d_in[1];
    const float* pbfc   = (const float*)d_in[2];
    const float* pwproj = (const float*)d_in[3];
    const float* pbproj = (const float*)d_in[4];
    const float* cwfc0  = (const float*)d_in[5];
    const float* cbfc0  = (const float*)d_in[6];
    const float* cwpr0  = (const float*)d_in[7];
    const float* cbpr0  = (const float*)d_in[8];
    const float* cwfc1  = (const float*)d_in[9];
    const float* cbfc1  = (const float*)d_in[10];
    const float* cwpr1  = (const float*)d_in[11];
    const float* cbpr1  = (const float*)d_in[12];
    float* out = (float*)d_out;

    int* cnt     = (int*)d_ws;                       // 16 counters
    int* rowlist = (int*)((char*)d_ws + 256);        // E*B ints (512 KB)

    init_cnt_kernel<<<1, 32, 0, stream>>>(cnt);
    parent_kernel<<<B / 16, 256, 0, stream>>>(x, pwfc, pbfc, pwproj, pbproj,
                                              out, cnt, rowlist);
    child_kernel<C0><<<dim3(B / 16, E), 256, 0, stream>>>(
        x, cwfc0, cbfc0, cwpr0, cbpr0, cnt, rowlist, out + OFF_CL0, out + OFF_CP0);
    child_kernel<C1><<<dim3(B / 16, E), 256, 0, stream>>>(
        x, cwfc1, cbfc1, cwpr1, cbpr1, cnt, rowlist, out + OFF_CL1, out + OFF_CP1);
}